// GraphModule_59012850647686
// MI455X (gfx1250) — compile-verified
//
#include <hip/hip_runtime.h>

#define DDIM 64

typedef float v2f __attribute__((ext_vector_type(2)));
typedef float v8f __attribute__((ext_vector_type(8)));

// Native CDNA5 L2-side f32 atomic add, device scope, non-returning.
// Forced via inline asm so we never fall back to a CAS loop: the 3 propagation
// hops issue ~230M of these, all landing in the 192MB L2.
__device__ __forceinline__ void atomic_add_f32(float* p, float v) {
    asm volatile("global_atomic_add_f32 %0, %1, off scope:SCOPE_DEV"
                 :
                 : "v"(p), "v"(v)
                 : "memory");
}

__global__ void zero4_kernel(float4* __restrict__ p, int n4) {
    int t = blockIdx.x * blockDim.x + threadIdx.x;
    if (t < n4) p[t] = make_float4(0.f, 0.f, 0.f, 0.f);
}

__global__ void degree_kernel(const int* __restrict__ col, float* __restrict__ deg, int E) {
    int e = blockIdx.x * blockDim.x + threadIdx.x;
    if (e < E) atomic_add_f32(&deg[col[e]], 1.0f);
}

__global__ void dinv_kernel(float* __restrict__ deg, int N) {
    int i = blockIdx.x * blockDim.x + threadIdx.x;
    if (i < N) {
        float d = deg[i];
        deg[i] = (d > 0.f) ? (1.0f / sqrtf(d)) : 0.f;
    }
}

__global__ void edgew_kernel(const int* __restrict__ row, const int* __restrict__ col,
                             const float* __restrict__ dinv, float* __restrict__ ew, int E) {
    int e = blockIdx.x * blockDim.x + threadIdx.x;
    if (e < E) ew[e] = dinv[row[e]] * dinv[col[e]];
}

__global__ void initout_kernel(const float4* __restrict__ w, const float* __restrict__ alpha,
                               float4* __restrict__ out, int n4) {
    int t = blockIdx.x * blockDim.x + threadIdx.x;
    if (t < n4) {
        float a = alpha[0];            // uniform -> scalar load
        float4 v = w[t];
        out[t] = make_float4(a * v.x, a * v.y, a * v.z, a * v.w);
    }
}

// One thread per (edge, 4-float slab): 16 consecutive threads cover one edge's
// full 64-float row -> coalesced 256B gather; 4 native f32 atomics land in L2.
__global__ void scatter_kernel(const int* __restrict__ row, const int* __restrict__ col,
                               const float* __restrict__ ew, const float* __restrict__ x,
                               float* __restrict__ xn, int E) {
    int t = blockIdx.x * blockDim.x + threadIdx.x;
    if (t >= E * 16) return;
    int e = t >> 4;
    int p = t & 15;
    int r = row[e];
    int c = col[e];
    float w = ew[e];
    const float4 v = *(const float4*)(x + (size_t)r * DDIM + p * 4);
    float* dst = xn + (size_t)c * DDIM + p * 4;
    atomic_add_f32(dst + 0, w * v.x);
    atomic_add_f32(dst + 1, w * v.y);
    atomic_add_f32(dst + 2, w * v.z);
    atomic_add_f32(dst + 3, w * v.w);
}

__global__ void axpy_kernel(float4* __restrict__ out, const float4* __restrict__ xn,
                            const float* __restrict__ alpha, int k, int n4) {
    int t = blockIdx.x * blockDim.x + threadIdx.x;
    if (t < n4) {
        float a = alpha[k];            // uniform -> scalar load
        float4 o = out[t];
        float4 v = xn[t];
        out[t] = make_float4(o.x + a * v.x, o.y + a * v.y, o.z + a * v.z, o.w + a * v.w);
    }
}

// One wave32 scores 16 edges with V_WMMA_F32_16X16X4_F32 (full f32 precision):
//   A[m][k] = out[row[base+m]][k],  B[k][n] = out[col[base+n]][k]
// accumulated over 16 chunks of K=4; scores are the diagonal of the 16x16 C.
// ISA layout: lane(m,h) A-pair = A[m][4c+2h .. 4c+2h+1]; B-pair = B[4c+2h..][n].
__global__ __launch_bounds__(256) void score_wmma_kernel(
        const int* __restrict__ row, const int* __restrict__ col,
        const float* __restrict__ outv, float* __restrict__ score, int E) {
    int gtid = blockIdx.x * blockDim.x + threadIdx.x;
    int wave = gtid >> 5;
    int lane = threadIdx.x & 31;
    int base = wave * 16;
    if (base >= E) return;            // wave-uniform: EXEC stays all-1s for WMMA

    int m = lane & 15;                // edge slot within tile (A-row / B-col)
    int h = lane >> 4;                // half-wave -> K sub-pair
    int e = base + m;
    if (e >= E) e = E - 1;            // clamp (tail tile); duplicate work is harmless
    int r = row[e];
    int c = col[e];
    const float* pa = outv + (size_t)r * DDIM + 2 * h;
    const float* pb = outv + (size_t)c * DDIM + 2 * h;

    v8f acc = {0.f, 0.f, 0.f, 0.f, 0.f, 0.f, 0.f, 0.f};
#pragma unroll
    for (int ch = 0; ch < 16; ++ch) {
        v2f a = *(const v2f*)(pa + 4 * ch);
        v2f b = *(const v2f*)(pb + 4 * ch);
        acc = __builtin_amdgcn_wmma_f32_16x16x4_f32(
            /*neg_a=*/false, a, /*neg_b=*/false, b,
            /*c_mod=*/(short)0, acc, /*reuse_a=*/false, /*reuse_b=*/false);
    }

    // Diagonal C[d][d]: lanes 0..7 hold d=lane at acc[lane];
    // lanes 24..31 hold d=lane-16 at acc[lane-24].
    int v = -1, d = -1;
    if (lane < 8)        { v = lane;      d = lane; }
    else if (lane >= 24) { v = lane - 24; d = lane - 16; }
    if (v >= 0) {
        float s = 0.f;
#pragma unroll
        for (int i = 0; i < 8; ++i)
            if (v == i) s = acc[i];
        if (base + d < E) score[base + d] = s;
    }
}

static inline int cdiv_i(long long a, long long b) { return (int)((a + b - 1) / b); }
static inline size_t align256(size_t x) { return (x + 255) & ~(size_t)255; }

extern "C" void kernel_launch(void* const* d_in, const int* in_sizes, int n_in,
                              void* d_out, int out_size, void* d_ws, size_t ws_size,
                              hipStream_t stream) {
    (void)n_in; (void)out_size; (void)ws_size;

    const int*   edge_index = (const int*)d_in[0];   // [2,E] flat: row then col
    const float* weight     = (const float*)d_in[1]; // [N,64]
    const float* alpha      = (const float*)d_in[2]; // [K+1]

    const int E = in_sizes[0] / 2;
    const int N = in_sizes[1] / DDIM;
    const int K = in_sizes[2] - 1;

    const int* rowp = edge_index;
    const int* colp = edge_index + E;
    float* score = (float*)d_out;

    // Workspace carve-up (~82 MB; whole working set is L2-resident on MI455X)
    char* ws = (char*)d_ws;
    size_t off = 0;
    float* deg = (float*)(ws + off); off += align256((size_t)N * 4);
    float* ew  = (float*)(ws + off); off += align256((size_t)E * 4);
    float* x0  = (float*)(ws + off); off += align256((size_t)N * DDIM * 4);
    float* x1  = (float*)(ws + off); off += align256((size_t)N * DDIM * 4);
    float* out = (float*)(ws + off); off += align256((size_t)N * DDIM * 4);

    const int B = 256;
    const int n4nd = N * DDIM / 4;

    // 1) degree -> dinv (in place)
    zero4_kernel<<<cdiv_i((N + 3) / 4, B), B, 0, stream>>>((float4*)deg, (N + 3) / 4);
    degree_kernel<<<cdiv_i(E, B), B, 0, stream>>>(colp, deg, E);
    dinv_kernel<<<cdiv_i(N, B), B, 0, stream>>>(deg, N);

    // 2) per-edge symmetric-norm weights (computed once, reused K times)
    edgew_kernel<<<cdiv_i(E, B), B, 0, stream>>>(rowp, colp, deg, ew, E);

    // 3) out = alpha[0] * weight
    initout_kernel<<<cdiv_i(n4nd, B), B, 0, stream>>>((const float4*)weight, alpha,
                                                      (float4*)out, n4nd);

    // 4) K propagation hops: x_{k} = scatter(ew * x_{k-1}[row] -> col); out += alpha[k]*x_k
    const float* xc = weight;
    float* bufs[2] = {x0, x1};
    for (int k = 1; k <= K; ++k) {
        float* xn = bufs[(k - 1) & 1];
        zero4_kernel<<<cdiv_i(n4nd, B), B, 0, stream>>>((float4*)xn, n4nd);
        scatter_kernel<<<cdiv_i((long long)E * 16, B), B, 0, stream>>>(rowp, colp, ew, xc, xn, E);
        axpy_kernel<<<cdiv_i(n4nd, B), B, 0, stream>>>((float4*)out, (const float4*)xn,
                                                       alpha, k, n4nd);
        xc = xn;
    }

    // 5) edge scores via f32 WMMA: one wave per 16 edges
    {
        long long waves = (E + 15) / 16;
        long long threads = waves * 32;
        score_wmma_kernel<<<cdiv_i(threads, B), B, 0, stream>>>(rowp, colp, out, score, E);
    }
}